// Quantize_19765439496211
// MI455X (gfx1250) — compile-verified
//
#include <hip/hip_runtime.h>
#include <hip/hip_bf16.h>
#include <stdint.h>

typedef __attribute__((ext_vector_type(2))) float v2f;
typedef __attribute__((ext_vector_type(8))) float v8f;
typedef unsigned int u32x4 __attribute__((ext_vector_type(4)));
typedef int i32x8 __attribute__((ext_vector_type(8)));
typedef int i32x4 __attribute__((ext_vector_type(4)));

#define N_ROWS 32768
#define KCODES 8192
#define DDIM   512

#if __has_builtin(__builtin_amdgcn_tensor_load_to_lds)
#define HAVE_TDM 1
#else
#define HAVE_TDM 0
#endif

__device__ __forceinline__ v8f wmma_f32(v2f a, v2f b, v8f c) {
  // V_WMMA_F32_16X16X4_F32: (neg_a, A, neg_b, B, c_mod, C, reuse_a, reuse_b)
  return __builtin_amdgcn_wmma_f32_16x16x4_f32(false, a, false, b, (short)0, c,
                                               false, false);
}

// ---------------------------------------------------------------------------
// codebook[c][j] = sum_i embed_w[c][i] * proj_w[j][i] + proj_b[j]
// One wave per 16x16 output tile; K-loop of 128 WMMA(16x16x4) steps.
// ---------------------------------------------------------------------------
__global__ __launch_bounds__(128) void k_codebook(const float* __restrict__ ew,
                                                  const float* __restrict__ pw,
                                                  const float* __restrict__ pb,
                                                  float* __restrict__ cb) {
  const int wave = (blockIdx.x * 128 + threadIdx.x) >> 5;
  const int lane = threadIdx.x & 31;
  const int half = lane >> 4, l = lane & 15;
  const int ct = wave >> 5;   // 0..511 : code tile
  const int jt = wave & 31;   // 0..31  : dim tile
  const float* arow = ew + (size_t)(ct * 16 + l) * DDIM + 2 * half;
  const float* brow = pw + (size_t)(jt * 16 + l) * DDIM + 2 * half;
  v8f acc = {};
  for (int i = 0; i < DDIM; i += 4) {
    v2f a = *(const v2f*)(arow + i);
    v2f b = *(const v2f*)(brow + i);
    acc = wmma_f32(a, b, acc);
  }
  const int col = jt * 16 + l;
  const float bias = pb[col];
#pragma unroll
  for (int v = 0; v < 8; ++v) {
    const int r = ct * 16 + v + 8 * half;
    cb[(size_t)r * DDIM + col] = acc[v] + bias;
  }
}

// ---------------------------------------------------------------------------
// dst[row] = ||src[row]||^2   (one wave per row)
// ---------------------------------------------------------------------------
__global__ __launch_bounds__(256) void k_rownorm(const float* __restrict__ src,
                                                 float* __restrict__ dst,
                                                 int rows) {
  const int row = blockIdx.x * 8 + (threadIdx.x >> 5);
  const int lane = threadIdx.x & 31;
  if (row >= rows) return;
  const float* p = src + (size_t)row * DDIM;
  float s = 0.f;
  for (int t = lane; t < DDIM; t += 32) { float v = p[t]; s += v * v; }
#pragma unroll
  for (int off = 16; off > 0; off >>= 1) s += __shfl_xor(s, off, 32);
  if (lane == 0) dst[row] = s;
}

// ---------------------------------------------------------------------------
// Main: d[n][k] = xnorm[n] + cnorm[k] - 2 * x[n].c[k], fused running argmin.
// Block = 256 threads (8 wave32) owns 64 rows; codes streamed in 64-wide
// chunks, double-buffered in LDS (2 x 132 KB of the 320 KB/WGP).
// Chunk staging: TENSOR_LOAD_TO_LDS (TDM, TENSORcnt) issued by wave 0 — one
// instruction DMAs the 64x512 f32 tile; tile_dim0=516 vs tensor_dim0=512
// zero-pads 4 floats/row in LDS (bank-conflict-free 516-f32 pitch).
// Fallback: GLOBAL_LOAD_ASYNC_TO_LDS_B128 (ASYNCcnt).
// Wave w: row tile (w&3)*16, code sub-tile (w>>2)*32 (two 16x16 WMMA accs).
// ---------------------------------------------------------------------------
#define KC  64
#define LDC 516

#if HAVE_TDM
__device__ __forceinline__ void tdm_fill_chunk(const float* __restrict__ gsrc,
                                               uint32_t lds_base_bytes) {
  const uint64_t ga = (uint64_t)(uintptr_t)gsrc;
  // D# group 0: count=1 | lds_addr | global_addr[56:0] | type=2
  u32x4 g0;
  g0.x = 1u;
  g0.y = lds_base_bytes;
  g0.z = (unsigned int)(ga & 0xffffffffu);
  g0.w = ((unsigned int)(ga >> 32) & 0x01ffffffu) | (2u << 30);
  // D# group 1: data_size=4B; tensor 512x64 (stride 512); tile 516x64
  i32x8 g1;
  g1[0] = (int)(2u << 16);            // data_size=2 (4 bytes)
  g1[1] = (int)(512u << 16);          // tensor_dim0[15:0] @ bits 63:48
  g1[2] = (int)(64u << 16);           // dim0 hi=0 | tensor_dim1[15:0]=64
  g1[3] = (int)(516u << 16);          // dim1 hi=0 | tile_dim0=516 (OOB pad)
  g1[4] = (int)64u;                   // tile_dim1=64, tile_dim2=0
  g1[5] = (int)512u;                  // tensor_dim0_stride low 32
  g1[6] = 0;
  g1[7] = 0;
  i32x4 gz4 = {0, 0, 0, 0};           // groups 2/3 unused (<=2D tensor)
  i32x8 gz8 = {0, 0, 0, 0, 0, 0, 0, 0};
  // amdgpu-toolchain 6-arg form: (g0, g1, g2, g3, g4, cpol)
  __builtin_amdgcn_tensor_load_to_lds(g0, g1, gz4, gz4, gz8, 0);
}
#else
__device__ __forceinline__ void async_fill_chunk(const float* __restrict__ gsrc,
                                                 uint32_t lds_base_bytes,
                                                 int tid) {
#pragma unroll
  for (int q = 0; q < (KC * DDIM / 4) / 256; ++q) {
    const int f = tid + 256 * q;            // float4 index
    const int code = f >> 7;                // / (DDIM/4)
    const int col = (f & 127) << 2;
    uint64_t ga = (uint64_t)(uintptr_t)(gsrc + (size_t)code * DDIM + col);
    uint32_t la = lds_base_bytes + (uint32_t)(code * LDC + col) * 4u;
    asm volatile("global_load_async_to_lds_b128 %0, %1, off"
                 :: "v"(la), "v"(ga) : "memory");
  }
}
#endif

__global__ __launch_bounds__(256) void k_main(const float* __restrict__ x,
                                              const float* __restrict__ cb,
                                              const float* __restrict__ xnorm,
                                              const float* __restrict__ cnorm,
                                              float* __restrict__ dout,
                                              int* __restrict__ argidx) {
  constexpr int BM = 64;
  __shared__ float ldsC[2][KC * LDC];    // 2 x 132 KB double buffer
  __shared__ float redV[2][BM];
  __shared__ int   redI[2][BM];

  const int tid = threadIdx.x, lane = tid & 31, wave = tid >> 5;
  const int half = lane >> 4, l = lane & 15;
  const int mt = wave & 3, cs = wave >> 2;
  const int m0 = blockIdx.x * BM;

  const uint32_t lbase[2] = {(uint32_t)(uintptr_t)&ldsC[0][0],
                             (uint32_t)(uintptr_t)&ldsC[1][0]};

  const float* xrow = x + (size_t)(m0 + mt * 16 + l) * DDIM + 2 * half;
  float xn[8];
#pragma unroll
  for (int v = 0; v < 8; ++v) xn[v] = xnorm[m0 + mt * 16 + v + 8 * half];

  float minV[8]; int minI[8];
#pragma unroll
  for (int v = 0; v < 8; ++v) { minV[v] = 3.4e38f; minI[v] = 0; }

  // prologue: fill buffer 0 with chunk 0
#if HAVE_TDM
  if (wave == 0) {
    tdm_fill_chunk(cb, lbase[0]);
    __builtin_amdgcn_s_wait_tensorcnt(0);
  }
#else
  async_fill_chunk(cb, lbase[0], tid);
  asm volatile("s_wait_asynccnt 0x0" ::: "memory");
#endif
  __syncthreads();

  int cur = 0;
  for (int c0 = 0; c0 < KCODES; c0 += KC) {
    // kick DMA of the next chunk into the other buffer (overlaps compute)
    if (c0 + KC < KCODES) {
#if HAVE_TDM
      if (wave == 0)
        tdm_fill_chunk(cb + (size_t)(c0 + KC) * DDIM, lbase[cur ^ 1]);
#else
      async_fill_chunk(cb + (size_t)(c0 + KC) * DDIM, lbase[cur ^ 1], tid);
#endif
    }

    v8f acc0 = {}, acc1 = {};
    const float* base = &ldsC[cur][0];
    const float* b0p = base + (cs * 32 + l) * LDC + 2 * half;
    const float* b1p = base + (cs * 32 + 16 + l) * LDC + 2 * half;
#pragma unroll 4
    for (int kk = 0; kk < DDIM; kk += 4) {
      v2f a  = *(const v2f*)(xrow + kk);
      v2f b0 = *(const v2f*)(b0p + kk);
      v2f b1 = *(const v2f*)(b1p + kk);
      acc0 = wmma_f32(a, b0, acc0);
      acc1 = wmma_f32(a, b1, acc1);
    }

    const int cbase = c0 + cs * 32;
    const float cn0 = cnorm[cbase + l];
    const float cn1 = cnorm[cbase + 16 + l];
#pragma unroll
    for (int v = 0; v < 8; ++v) {
      const int r = m0 + mt * 16 + v + 8 * half;
      const float d0 = xn[v] + cn0 - 2.0f * acc0[v];
      const float d1 = xn[v] + cn1 - 2.0f * acc1[v];
      dout[(size_t)r * KCODES + cbase + l]      = d0;
      dout[(size_t)r * KCODES + cbase + 16 + l] = d1;
      // strict '<' + ascending code order => first-occurrence argmin
      if (d0 < minV[v]) { minV[v] = d0; minI[v] = cbase + l; }
      if (d1 < minV[v]) { minV[v] = d1; minI[v] = cbase + 16 + l; }
    }

    // next-chunk DMA complete + all waves done reading `cur`
#if HAVE_TDM
    if (wave == 0) __builtin_amdgcn_s_wait_tensorcnt(0);
#else
    asm volatile("s_wait_asynccnt 0x0" ::: "memory");
#endif
    __syncthreads();
    cur ^= 1;
  }

  // reduce (min,idx) across the 16 lanes of each half-wave
#pragma unroll
  for (int off = 1; off < 16; off <<= 1) {
#pragma unroll
    for (int v = 0; v < 8; ++v) {
      float ov = __shfl_xor(minV[v], off, 32);
      int   oi = __shfl_xor(minI[v], off, 32);
      if (ov < minV[v] || (ov == minV[v] && oi < minI[v])) {
        minV[v] = ov; minI[v] = oi;
      }
    }
  }
  if (l == 0) {  // lanes 0 and 16 hold rows v and v+8
#pragma unroll
    for (int v = 0; v < 8; ++v) {
      const int rl = mt * 16 + v + 8 * half;
      redV[cs][rl] = minV[v];
      redI[cs][rl] = minI[v];
    }
  }
  __syncthreads();
  if (tid < BM) {
    float v0 = redV[0][tid]; int i0 = redI[0][tid];
    float v1 = redV[1][tid]; int i1 = redI[1][tid];
    argidx[m0 + tid] = (v1 < v0 || (v1 == v0 && i1 < i0)) ? i1 : i0;
  }
}

// ---------------------------------------------------------------------------
// z_q[row] = codebook[idx[row]]; rowsq[row] = sum((z_q - x)^2); ind as float
// ---------------------------------------------------------------------------
__global__ __launch_bounds__(256) void k_gather(const float* __restrict__ x,
                                                const float* __restrict__ cb,
                                                const int* __restrict__ argidx,
                                                float* __restrict__ zq,
                                                float* __restrict__ ind_out,
                                                float* __restrict__ rowsq) {
  const int row = blockIdx.x * 8 + (threadIdx.x >> 5);
  const int lane = threadIdx.x & 31;
  const int idx = argidx[row];
  const float* crow = cb + (size_t)idx * DDIM;
  const float* xr = x + (size_t)row * DDIM;
  float s = 0.f;
  for (int t = lane; t < DDIM; t += 32) {
    const float c = crow[t];
    const float d = c - xr[t];
    zq[(size_t)row * DDIM + t] = c;
    s += d * d;
  }
#pragma unroll
  for (int off = 16; off > 0; off >>= 1) s += __shfl_xor(s, off, 32);
  if (lane == 0) { rowsq[row] = s; ind_out[row] = (float)idx; }
}

// deterministic final reduction: diff = 1.25 * mean(sq)
__global__ __launch_bounds__(256) void k_reduce(const float* __restrict__ rowsq,
                                                float* __restrict__ diff_out) {
  __shared__ float sh[256];
  float s = 0.f;
  for (int i = 0; i < N_ROWS / 256; ++i) s += rowsq[threadIdx.x + 256 * i];
  sh[threadIdx.x] = s;
  __syncthreads();
  for (int stride = 128; stride > 0; stride >>= 1) {
    if (threadIdx.x < stride) sh[threadIdx.x] += sh[threadIdx.x + stride];
    __syncthreads();
  }
  if (threadIdx.x == 0)
    diff_out[0] = sh[0] * (1.25f / 16777216.0f);  // N_ROWS*DDIM = 16777216
}

extern "C" void kernel_launch(void* const* d_in, const int* in_sizes, int n_in,
                              void* d_out, int out_size, void* d_ws,
                              size_t ws_size, hipStream_t stream) {
  (void)in_sizes; (void)n_in; (void)out_size; (void)ws_size;
  const float* x  = (const float*)d_in[0];   // [32768, 512]
  const float* ew = (const float*)d_in[1];   // [8192, 512]
  const float* pw = (const float*)d_in[2];   // [512, 512]
  const float* pb = (const float*)d_in[3];   // [512]
  // d_in[4] = is_look_back (slice is a no-op since all tokens elected)

  float* ws    = (float*)d_ws;
  float* cbp   = ws;                               // 8192*512
  float* cnorm = cbp + (size_t)KCODES * DDIM;      // 8192
  float* xnormp= cnorm + KCODES;                   // 32768
  float* rowsq = xnormp + N_ROWS;                  // 32768
  int*   argix = (int*)(rowsq + N_ROWS);           // 32768

  float* out  = (float*)d_out;
  float* zq   = out;                               // 16777216
  float* diff = out + (size_t)N_ROWS * DDIM;       // 1
  float* ind  = diff + 1;                          // 32768
  float* dmat = ind + N_ROWS;                      // 32768*8192

  k_codebook<<<dim3((KCODES / 16) * (DDIM / 16) / 4), dim3(128), 0, stream>>>(
      ew, pw, pb, cbp);
  k_rownorm<<<dim3(KCODES / 8), dim3(256), 0, stream>>>(cbp, cnorm, KCODES);
  k_rownorm<<<dim3(N_ROWS / 8), dim3(256), 0, stream>>>(x, xnormp, N_ROWS);
  k_main<<<dim3(N_ROWS / 64), dim3(256), 0, stream>>>(x, cbp, xnormp, cnorm,
                                                      dmat, argix);
  k_gather<<<dim3(N_ROWS / 8), dim3(256), 0, stream>>>(x, cbp, argix, zq, ind,
                                                       rowsq);
  k_reduce<<<dim3(1), dim3(256), 0, stream>>>(rowsq, diff);
}